// Self_Attention_SMC_4930622456370
// MI455X (gfx1250) — compile-verified
//
#include <hip/hip_runtime.h>
#include <hip/hip_bf16.h>

typedef float v2f __attribute__((ext_vector_type(2)));
typedef float v4f __attribute__((ext_vector_type(4)));
typedef float v8f __attribute__((ext_vector_type(8)));

#define BP   256      // B*P
#define S    1024
#define D    256
#define KVN  (BP * S * D)   // 67,108,864 elements per cache

// ---------------------------------------------------------------------------
// Y(256x256) = X(256x256) @ W(256x256) + bias, via V_WMMA_F32_16X16X4_F32.
// One wave per 16x16 output tile; 8 waves (256 threads) per block; 32 blocks.
// ---------------------------------------------------------------------------
__global__ __launch_bounds__(256)
void gemm256_bias_wmma(const float* __restrict__ X,
                       const float* __restrict__ W,
                       const float* __restrict__ bias,
                       float* __restrict__ Y)
{
    const int lane = threadIdx.x & 31;
    const int wave = threadIdx.x >> 5;
    const int tid  = blockIdx.x * 8 + wave;     // 0..255 tiles
    const int tileM = tid >> 4;
    const int tileN = tid & 15;

    const int l    = lane & 15;   // row (A) / col (B,C)
    const int half = lane >> 4;   // K-half selector

    const float* __restrict__ xrow = X + (size_t)(tileM * 16 + l) * D;

    v8f acc = {};
#pragma unroll 4
    for (int k0 = 0; k0 < D; k0 += 4) {
        const int ka = k0 + 2 * half;
        // A 16x4 fp32: lanes 0-15 -> K = k0,k0+1 ; lanes 16-31 -> K = k0+2,k0+3
        v2f a;
        a.x = xrow[ka];
        a.y = xrow[ka + 1];
        // B 4x16 fp32: lane l -> N, same K-half split as A
        const float* __restrict__ wp = W + (size_t)ka * D + tileN * 16 + l;
        v2f b;
        b.x = wp[0];
        b.y = wp[D];
        acc = __builtin_amdgcn_wmma_f32_16x16x4_f32(
            /*neg_a=*/false, a, /*neg_b=*/false, b,
            /*c_mod=*/(short)0, acc, /*reuse_a=*/false, /*reuse_b=*/false);
    }

    const int n = tileN * 16 + l;
    const float bv = bias[n];
#pragma unroll
    for (int r = 0; r < 8; ++r) {
        const int m = tileM * 16 + r + 8 * half;   // C/D: VGPR r -> M=r (+8 for hi half)
        Y[(size_t)m * D + n] = acc[r] + bv;
    }
}

// ---------------------------------------------------------------------------
// Fused per-(b,p) attention: splice k into K -> K_new while computing logits,
// softmax -> attn, splice v into V -> V_new while accumulating z_pre = p @ V.
// One 256-thread block per (b,p). K/V each streamed exactly once, with
// non-temporal b128 loads/stores (read-once / write-once, keep L2 clean).
// ---------------------------------------------------------------------------
__global__ __launch_bounds__(256)
void attn_fused(const float* __restrict__ K,
                const float* __restrict__ V,
                const float* __restrict__ qproj,
                const float* __restrict__ kproj,
                const float* __restrict__ vproj,
                const int*   __restrict__ timestep_p,
                float* __restrict__ K_new,
                float* __restrict__ V_new,
                float* __restrict__ attn,
                float* __restrict__ zpre)
{
    const int bp = blockIdx.x;
    const int t  = threadIdx.x;
    const int ts = *timestep_p;

    __shared__ float q_s[D];
    __shared__ float lg[S];
    __shared__ float red[256];

    const float* __restrict__ Kbase = K     + (size_t)bp * S * D;
    float*       __restrict__ Kout  = K_new + (size_t)bp * S * D;
    const float* __restrict__ Vbase = V     + (size_t)bp * S * D;
    float*       __restrict__ Vout  = V_new + (size_t)bp * S * D;
    const float* __restrict__ qrow  = qproj + (size_t)bp * D;
    const float* __restrict__ krow  = kproj + (size_t)bp * D;
    const float* __restrict__ vrow  = vproj + (size_t)bp * D;

    q_s[t] = qrow[t];
    __syncthreads();

    // ---- K pass: copy/splice + logits. 8 lanes per row, 32 rows per pass.
    const int grp   = t >> 3;   // row-in-pass 0..31
    const int lane8 = t & 7;    // segment lane 0..7
    for (int pass = 0; pass < S / 32; ++pass) {
        const int s = pass * 32 + grp;
        const float* __restrict__ src = (s == ts) ? krow : (Kbase + (size_t)s * D);
        float*       __restrict__ dst = Kout + (size_t)s * D;
        float acc = 0.0f;
#pragma unroll
        for (int it = 0; it < 8; ++it) {
            const int off = it * 32 + lane8 * 4;
            const v4f v = __builtin_nontemporal_load((const v4f*)(src + off));
            __builtin_nontemporal_store(v, (v4f*)(dst + off));
            acc += v.x * q_s[off] + v.y * q_s[off + 1] +
                   v.z * q_s[off + 2] + v.w * q_s[off + 3];
        }
        // reduce the 8 partial sums (lanes of a group are contiguous in-wave)
        acc += __shfl_xor(acc, 1, 32);
        acc += __shfl_xor(acc, 2, 32);
        acc += __shfl_xor(acc, 4, 32);
        if (lane8 == 0) lg[s] = acc * 0.0625f;   // * 1/sqrt(256)
    }
    __syncthreads();

    // ---- softmax over S=1024 logits
    float m = -3.0e38f;
    for (int i = t; i < S; i += 256) m = fmaxf(m, lg[i]);
    red[t] = m;
    __syncthreads();
    for (int off = 128; off > 0; off >>= 1) {
        if (t < off) red[t] = fmaxf(red[t], red[t + off]);
        __syncthreads();
    }
    const float mx = red[0];
    __syncthreads();
    float lsum = 0.0f;
    for (int i = t; i < S; i += 256) {
        const float e = __expf(lg[i] - mx);
        lg[i] = e;
        lsum += e;
    }
    red[t] = lsum;
    __syncthreads();
    for (int off = 128; off > 0; off >>= 1) {
        if (t < off) red[t] += red[t + off];
        __syncthreads();
    }
    const float inv = 1.0f / red[0];
    for (int i = t; i < S; i += 256) {
        const float pn = lg[i] * inv;
        lg[i] = pn;
        attn[(size_t)bp * S + i] = pn;
    }
    __syncthreads();

    // ---- V pass: copy/splice + z_pre[d] = sum_s p[s] * V_new[s][d]
    // 4 row-groups x 64 threads; each thread owns 4 contiguous columns (b128).
    const int rgrp = t >> 6;          // s % 4 handled by this thread
    const int col  = (t & 63) * 4;    // column quad
    const v4f vnew4 = *(const v4f*)(vrow + col);
    v4f acc4 = {};
#pragma unroll 4
    for (int it = 0; it < S / 4; ++it) {
        const int s = it * 4 + rgrp;
        v4f val = __builtin_nontemporal_load((const v4f*)(Vbase + (size_t)s * D + col));
        val = (s == ts) ? vnew4 : val;
        __builtin_nontemporal_store(val, (v4f*)(Vout + (size_t)s * D + col));
        const float p = lg[s];
        acc4.x = fmaf(p, val.x, acc4.x);
        acc4.y = fmaf(p, val.y, acc4.y);
        acc4.z = fmaf(p, val.z, acc4.z);
        acc4.w = fmaf(p, val.w, acc4.w);
    }
    __syncthreads();                       // done reading lg; reuse as scratch
    *(v4f*)(&lg[rgrp * 256 + col]) = acc4; // partial[rgrp][col..col+3]
    __syncthreads();
    if (t < 64) {
        const int c = t * 4;
        const v4f p0 = *(const v4f*)(&lg[0 * 256 + c]);
        const v4f p1 = *(const v4f*)(&lg[1 * 256 + c]);
        const v4f p2 = *(const v4f*)(&lg[2 * 256 + c]);
        const v4f p3 = *(const v4f*)(&lg[3 * 256 + c]);
        const v4f zr = (p0 + p1) + (p2 + p3);
        *(v4f*)(zpre + (size_t)bp * D + c) = zr;
    }
}

// ---------------------------------------------------------------------------
extern "C" void kernel_launch(void* const* d_in, const int* in_sizes, int n_in,
                              void* d_out, int out_size, void* d_ws, size_t ws_size,
                              hipStream_t stream)
{
    (void)in_sizes; (void)n_in; (void)out_size; (void)ws_size;

    const float* inputs = (const float*)d_in[0];
    const int*   ts     = (const int*)  d_in[1];
    const float* K      = (const float*)d_in[2];
    const float* V      = (const float*)d_in[3];
    const float* Wq     = (const float*)d_in[4];
    const float* bq     = (const float*)d_in[5];
    const float* Wk     = (const float*)d_in[6];
    const float* bk     = (const float*)d_in[7];
    const float* Wv     = (const float*)d_in[8];
    const float* bv     = (const float*)d_in[9];
    const float* Wz     = (const float*)d_in[10];
    const float* bz     = (const float*)d_in[11];

    // outputs concatenated flat: z | K_new | V_new | attn
    float* out   = (float*)d_out;
    float* z     = out;
    float* K_new = z + (size_t)BP * D;
    float* V_new = K_new + (size_t)KVN;
    float* attn  = V_new + (size_t)KVN;

    // workspace: q | k | v | z_pre  (each 256*256 floats = 1 MiB total)
    float* ws = (float*)d_ws;
    float* q  = ws;
    float* kp = ws + (size_t)BP * D;
    float* vp = ws + 2 * (size_t)BP * D;
    float* zp = ws + 3 * (size_t)BP * D;

    gemm256_bias_wmma<<<32, 256, 0, stream>>>(inputs, Wq, bq, q);
    gemm256_bias_wmma<<<32, 256, 0, stream>>>(inputs, Wk, bk, kp);
    gemm256_bias_wmma<<<32, 256, 0, stream>>>(inputs, Wv, bv, vp);

    attn_fused<<<BP, 256, 0, stream>>>(K, V, q, kp, vp, ts, K_new, V_new, attn, zp);

    gemm256_bias_wmma<<<32, 256, 0, stream>>>(zp, Wz, bz, z);
}